// LocationEffect_68367289417954
// MI455X (gfx1250) — compile-verified
//
#include <hip/hip_runtime.h>
#include <hip/hip_bf16.h>
#include <math.h>

typedef __attribute__((ext_vector_type(2))) float v2f;
typedef __attribute__((ext_vector_type(4))) float v4f;
typedef __attribute__((ext_vector_type(8))) float v8f;
typedef __attribute__((ext_vector_type(4))) int   v4i;

#define TT 12
#define NN 1000
#define DD 64
#define NEGV (-1.0e12f)
#define NTILE 63          // ceil(1000/16) -> covers columns/rows 0..1007
#define NPAD 1008
#define SSTR 1020         // LDS row stride (floats); 1020%64==60 -> conflict-free column reads

// ---------------------------------------------------------------------------
// Stage 1: adjacency-masked softmax attention per timestamp (GAT), f32 WMMA.
// One block = one 16-row tile of one timestamp. 4 waves / 128 threads.
// Branchless: out-of-range rows are address-clamped; their results are either
// masked to NEG (pass 1) or multiplied by exactly-zero attention (pass 3).
// ---------------------------------------------------------------------------
__global__ __launch_bounds__(128) void gat_kernel(const float* __restrict__ h,
                                                  const int* __restrict__ adj,
                                                  float* __restrict__ nf) {
    extern __shared__ float smem[];
    float* sc     = smem;               // [16][SSTR] masked scaled scores -> exp values
    float* rowinv = smem + 16 * SSTR;   // [16] 1/rowsum

    const int itile = blockIdx.x;       // row tile 0..62
    const int t     = blockIdx.y;       // timestamp
    const int tid   = threadIdx.x;
    const int wave  = tid >> 5;
    const int lane  = tid & 31;
    const int lrow  = lane & 15;        // row (A) / col (B,C) index within tile
    const int hi    = lane >> 4;        // lane half selects K/M sub-block

    const float* ht = h + (size_t)t * NN * DD;

    // --- preload Q tile (16x64) in WMMA-A layout: lane holds row lrow,
    //     chunk c covers K = 4c + 2*hi + {0,1}. Clamp invalid rows (masked later).
    const int qrow  = itile * 16 + lrow;
    const int qrowc = qrow < NN ? qrow : NN - 1;
    v2f q[16];
    {
        const float* qp = ht + (size_t)qrowc * DD + 2 * hi;
        #pragma unroll
        for (int c = 0; c < 16; ++c) { q[c].x = qp[4 * c]; q[c].y = qp[4 * c + 1]; }
    }

    // --- Pass 1: S = Q*K^T over all column tiles; mask + scale into LDS
    for (int j = wave; j < NTILE; j += 4) {
        const int krow  = j * 16 + lrow;
        const int krowc = krow < NN ? krow : NN - 1;    // clamp; col>=NN masked below
        v2f kv[16];
        {
            const float* kp = ht + (size_t)krowc * DD + 2 * hi;
            #pragma unroll
            for (int c = 0; c < 16; ++c) { kv[c].x = kp[4 * c]; kv[c].y = kp[4 * c + 1]; }
        }
        const int col    = j * 16 + lrow;               // C-layout: lane holds column lrow
        const bool colok = col < NN;
        const int colc   = colok ? col : 0;
        int av[8];
        #pragma unroll
        for (int r = 0; r < 8; ++r) {                   // adj mask loads, pre-WMMA
            const int nrow = itile * 16 + (r + 8 * hi);
            const int nrc  = nrow < NN ? nrow : 0;
            av[r] = adj[(size_t)nrc * NN + colc];
        }
        v8f acc = {};
        #pragma unroll
        for (int c = 0; c < 16; ++c)
            acc = __builtin_amdgcn_wmma_f32_16x16x4_f32(false, q[c], false, kv[c],
                                                        (short)0, acc, false, false);
        // scheduling hint: batch all VMEM reads, then the WMMA chain
        __builtin_amdgcn_sched_group_barrier(0x020, 24, 0);  // VMEM read x24
        __builtin_amdgcn_sched_group_barrier(0x008, 16, 0);  // WMMA x16
        #pragma unroll
        for (int r = 0; r < 8; ++r) {
            const int m    = r + 8 * hi;                // row within tile
            const int nrow = itile * 16 + m;
            const bool keep = (nrow < NN) & colok & (av[r] > 0);
            sc[m * SSTR + col] = keep ? acc[r] * 0.125f : NEGV;
        }
    }
    __syncthreads();

    // --- Pass 2: row softmax (exp stored in LDS; store 1/sum, exact divide once)
    {
        const int row = tid >> 3;       // 16 rows x 8 threads
        const int sub = tid & 7;
        float* sr = sc + row * SSTR;
        const int c0 = sub * (NPAD / 8);
        float mx = -3.0e38f;
        for (int c = c0; c < c0 + NPAD / 8; ++c) mx = fmaxf(mx, sr[c]);
        mx = fmaxf(mx, __shfl_xor(mx, 1));
        mx = fmaxf(mx, __shfl_xor(mx, 2));
        mx = fmaxf(mx, __shfl_xor(mx, 4));
        float s = 0.f;
        for (int c = c0; c < c0 + NPAD / 8; ++c) {
            float e = __expf(sr[c] - mx);
            sr[c] = e;
            s += e;
        }
        s += __shfl_xor(s, 1);
        s += __shfl_xor(s, 2);
        s += __shfl_xor(s, 4);
        if (sub == 0) rowinv[row] = 1.0f / s;
    }
    __syncthreads();

    // --- Pass 3: O = E * H ; wave w owns output d-columns [16w, 16w+16).
    // B rows >= NN are clamped: the matching E columns are exactly 0.
    const int d0 = wave * 16;
    v8f o = {};
    #pragma unroll 4
    for (int k0 = 0; k0 < NPAD; k0 += 4) {
        v2f a = *(const v2f*)(sc + lrow * SSTR + k0 + 2 * hi);   // A: attn rows
        const int hr  = k0 + 2 * hi;                             // B: h rows hr, hr+1
        const int hrc = hr < NN - 1 ? hr : NN - 2;               // clamp (E col == 0 there)
        v2f b;
        b.x = ht[(size_t)hrc * DD + d0 + lrow];
        b.y = ht[(size_t)(hrc + 1) * DD + d0 + lrow];
        o = __builtin_amdgcn_wmma_f32_16x16x4_f32(false, a, false, b,
                                                  (short)0, o, false, false);
    }
    #pragma unroll
    for (int r = 0; r < 8; ++r) {
        const int m    = r + 8 * hi;
        const int nrow = itile * 16 + m;
        if (nrow < NN)
            nf[((size_t)t * NN + nrow) * DD + d0 + lrow] = o[r] * rowinv[m];
    }
}

// ---------------------------------------------------------------------------
// Stage 2: per-node temporal sigmoid gram matrix, stored pre-transposed:
//   swp[s2][b][t1] = sigmoid( dot(nf[t1,b,:], nf[s2,b,:]) / 8 )
// ---------------------------------------------------------------------------
__global__ __launch_bounds__(192) void temporal_kernel(const float* __restrict__ nf,
                                                       float* __restrict__ swp) {
    __shared__ float s[TT * DD];
    const int b = blockIdx.x;
    for (int idx = threadIdx.x; idx < TT * DD; idx += blockDim.x) {
        const int t = idx >> 6, d = idx & 63;
        s[idx] = nf[((size_t)t * NN + b) * DD + d];
    }
    __syncthreads();
    const int p = threadIdx.x;
    if (p < TT * TT) {
        const int t1 = p / TT, s2 = p % TT;
        float acc = 0.f;
        #pragma unroll 8
        for (int d = 0; d < DD; ++d) acc += s[t1 * DD + d] * s[s2 * DD + d];
        acc *= 0.125f;
        swp[((size_t)s2 * NN + b) * TT + t1] = 1.0f / (1.0f + __expf(-acc));
    }
}

// ---------------------------------------------------------------------------
// Stage 3: out[(s2*N+b)*12000 + t1*1000 + a] = swp[s2,b,t1] * adj[b,a]
// 576 MB write-once stream: non-temporal b128 stores so the L2-resident adj
// (4 MB, reused 144x) is not evicted by the output stream.
// ---------------------------------------------------------------------------
__global__ __launch_bounds__(256) void expand_kernel(const float* __restrict__ swp,
                                                     const int* __restrict__ adj,
                                                     float* __restrict__ out) {
    const int b  = blockIdx.x;   // node
    const int s2 = blockIdx.y;   // output row-block timestamp
    const int t1 = blockIdx.z;   // output col-block timestamp
    const float scale = swp[((size_t)s2 * NN + b) * TT + t1];
    const int tid = threadIdx.x;
    if (tid < NN / 4) {
        const v4i a4 = *(const v4i*)(adj + (size_t)b * NN + tid * 4);
        v4f o;
        o.x = scale * (float)a4.x;
        o.y = scale * (float)a4.y;
        o.z = scale * (float)a4.z;
        o.w = scale * (float)a4.w;
        float* dst = out + ((size_t)(s2 * NN + b)) * (TT * NN) + (size_t)t1 * NN + tid * 4;
        __builtin_nontemporal_store(o, (v4f*)dst);
    }
}

extern "C" void kernel_launch(void* const* d_in, const int* in_sizes, int n_in,
                              void* d_out, int out_size, void* d_ws, size_t ws_size,
                              hipStream_t stream) {
    const float* h   = (const float*)d_in[0];   // [T,N,D] f32
    const int*   adj = (const int*)d_in[1];     // [N,N] i32
    float* out = (float*)d_out;                 // [12000,12000] f32
    float* nf  = (float*)d_ws;                  // [T,N,D] f32  (3.07 MB)
    float* swp = nf + (size_t)TT * NN * DD;     // [T,N,T] f32  (0.58 MB)

    const size_t smem = (size_t)(16 * SSTR + 16) * sizeof(float);  // ~65.3 KB
    gat_kernel<<<dim3(NTILE, TT), 128, smem, stream>>>(h, adj, nf);
    temporal_kernel<<<dim3(NN), 192, 0, stream>>>(nf, swp);
    expand_kernel<<<dim3(NN, TT, TT), 256, 0, stream>>>(swp, adj, out);
}